// quantized_CSI_feedback_77060303225009
// MI455X (gfx1250) — compile-verified
//
#include <hip/hip_runtime.h>

// ---------------- CDNA5 / gfx1250 WMMA types ----------------
typedef __attribute__((ext_vector_type(2))) float v2f;
typedef __attribute__((ext_vector_type(8))) float v8f;

#define NPOS   114688   // 32 batch * 14 time * 256 rb
#define NBT    448      // 32 * 14
#define NRB    256
#define NC     32

// beta = 4.4492*ER^2 + 4.5655*ER + 1.2982, ER = 0.25
#define BETA_F  (4.4492f*0.0625f + 4.5655f*0.25f + 1.2982f)

// codebook: c = 4*l + n ; w = 0.5*[1, e^{i*pi*l/4}, i^n, i^n*e^{i*pi*l/4}]
__device__ __forceinline__ void codebook_w(int c, float wr[4], float wi[4]) {
  int l = c >> 2, n = c & 3;
  float th = 0.78539816339744830962f * (float)l;
  float st, ct;
  sincosf(th, &st, &ct);
  float pr, pim;
  switch (n) {
    case 0:  pr = 1.f;  pim = 0.f;  break;
    case 1:  pr = 0.f;  pim = 1.f;  break;
    case 2:  pr = -1.f; pim = 0.f;  break;
    default: pr = 0.f;  pim = -1.f; break;
  }
  wr[0] = 0.5f;            wi[0] = 0.f;
  wr[1] = 0.5f * ct;       wi[1] = 0.5f * st;
  wr[2] = 0.5f * pr;       wi[2] = 0.5f * pim;
  wr[3] = 0.5f * (pr*ct - pim*st);
  wi[3] = 0.5f * (pr*st + pim*ct);
}

__global__ void k_zero(float* __restrict__ p, int n) {
  int i = blockIdx.x * blockDim.x + threadIdx.x;
  if (i < n) p[i] = 0.f;
}

__device__ __forceinline__ float sum12(const float* __restrict__ p) {
  // p is 16B-aligned (offset is a multiple of 12 floats = 48B): 3x b128 loads
  const float4* q = (const float4*)p;
  float4 a = q[0], b = q[1], c = q[2];
  return ((a.x + a.y) + (a.z + a.w))
       + ((b.x + b.y) + (b.z + b.w))
       + ((c.x + c.y) + (c.z + c.w));
}

// RB-average: h(32,1,2,1,4,14,3072) -> havg[pos][16], pos=(b*14+t)*256+rb
// comp layout per position: [Re h0,0..3 | Im h0,0..3 | Re h1,0..3 | Im h1,0..3]
__global__ void k_rb_avg(const float* __restrict__ hr,
                         const float* __restrict__ hi,
                         float* __restrict__ havg) {
  int i = blockIdx.x * blockDim.x + threadIdx.x;      // over (b,rx,tx,t,rb)
  if (i >= 32*2*4*14*256) return;
  int rb = i % 256;
  int t  = (i / 256) % 14;
  int tx = (i / (256*14)) % 4;
  int rx = (i / (256*14*4)) % 2;
  int b  =  i / (256*14*4*2);
  size_t base = (size_t)b*344064 + (size_t)rx*172032 + (size_t)tx*43008
              + (size_t)t*3072 + (size_t)rb*12;
  float sr = sum12(hr + base) * (1.f/12.f);
  float si = sum12(hi + base) * (1.f/12.f);
  size_t pos = (size_t)((b*14 + t)*256 + rb);
  havg[pos*16 + rx*8 + tx]     = sr;
  havg[pos*16 + rx*8 + 4 + tx] = si;
}

// R = sum over positions, rx of h_row^H h_row  (4x4 complex, interleaved re/im)
__global__ void k_rsum(const float* __restrict__ havg, float* __restrict__ R) {
  __shared__ float sR[32];
  int tid = threadIdx.x;
  if (tid < 32) sR[tid] = 0.f;
  __syncthreads();
  int pos = blockIdx.x * blockDim.x + tid;
  float loc[32];
  #pragma unroll
  for (int k = 0; k < 32; ++k) loc[k] = 0.f;
  if (pos < NPOS) {
    const float* v = havg + (size_t)pos * 16;
    #pragma unroll
    for (int r = 0; r < 2; ++r) {
      const float* re = v + r*8;
      const float* im = re + 4;
      #pragma unroll
      for (int i = 0; i < 4; ++i)
        #pragma unroll
        for (int j = 0; j < 4; ++j) {
          loc[(i*4+j)*2+0] += re[i]*re[j] + im[i]*im[j];   // Re conj(hi)*hj
          loc[(i*4+j)*2+1] += re[i]*im[j] - im[i]*re[j];   // Im conj(hi)*hj
        }
    }
  }
  #pragma unroll
  for (int k = 0; k < 32; ++k) atomicAdd(&sR[k], loc[k]);
  __syncthreads();
  if (tid < 32) atomicAdd(&R[tid], sR[tid]);
}

// nv[c] = Re(w_c^H R w_c) / (NPOS * SNR_LIN)
__global__ void k_nv(const float* __restrict__ R, float* __restrict__ nv) {
  int c = threadIdx.x;
  if (c >= NC) return;
  float wr[4], wi[4];
  codebook_w(c, wr, wi);
  float acc = 0.f;
  #pragma unroll
  for (int i = 0; i < 4; ++i)
    #pragma unroll
    for (int j = 0; j < 4; ++j) {
      float Rre = R[(i*4+j)*2], Rim = R[(i*4+j)*2+1];
      float ar = wr[i]*wr[j] + wi[i]*wi[j];   // conj(wi)*wj
      float ai = wr[i]*wi[j] - wi[i]*wr[j];
      acc += ar*Rre - ai*Rim;
    }
  nv[c] = acc / ((float)NPOS * 10.f);
}

__device__ __forceinline__ float sinr_expterm(float p, float x, float nv) {
  float px = p + x;                               // |a+b|^2
  float Q  = p*p + nv*px;                         // h^H M h  (real)
  float rp = p*p*p + 2.f*nv*px*(p + nv);          // h^H M M^H h
  float sinr = __fdividef(Q*Q, nv*rp);
  return expf(-sinr * (1.f/BETA_F));
}

// Per-RB rate via WMMA: one block (one wave32) per RB.
// A (16x16 per tile, 4 K-chunks): fixed map {16 channel comps}->{4 precoders x
// (Re a, Im a, Re b, Im b)}. B: 16 positions' channel vectors. 8 tiles -> 32 c.
__global__ void __launch_bounds__(32) k_rate(const float* __restrict__ havg,
                                             const float* __restrict__ nvg,
                                             int*   __restrict__ pmi,
                                             float* __restrict__ rsel,
                                             float* __restrict__ pmat) {
  int rb = blockIdx.x;
  int l  = threadIdx.x;
  int h  = l >> 4;      // lane half: selects K sub-pair (A/B) and row group (D)
  int r  = l & 15;      // A: row;  B/D: column (= position within group)
  int ty   = r & 3;     // output component: 0 Re a, 1 Im a, 2 Re b, 3 Im b
  int csub = r >> 2;    // precoder within tile for this A row

  // ---- constant A fragments: aF[tile j][K-chunk q], per documented layout:
  // lane m = M row (l&15), VGPR0/1 = K = 4q + 2h + {0,1}
  v2f aF[8][4];
  #pragma unroll
  for (int j = 0; j < 8; ++j) {
    float wr[4], wi[4];
    codebook_w(4*j + csub, wr, wi);
    #pragma unroll
    for (int q = 0; q < 4; ++q) {
      float e[2];
      #pragma unroll
      for (int ee = 0; ee < 2; ++ee) {
        int k = 4*q + 2*h + ee;
        float val = 0.f;
        if (ty == 0)      { if (k < 4) val =  wr[k];   else if (k < 8)  val = -wi[k-4]; }
        else if (ty == 1) { if (k < 4) val =  wi[k];   else if (k < 8)  val =  wr[k-4]; }
        else if (ty == 2) { if (k >= 8 && k < 12) val = wr[k-8];  else if (k >= 12) val = -wi[k-12]; }
        else              { if (k >= 8 && k < 12) val = wi[k-8];  else if (k >= 12) val =  wr[k-12]; }
        e[ee] = val;
      }
      v2f a; a.x = e[0]; a.y = e[1];
      aF[j][q] = a;
    }
  }

  // per-lane noise variances for the two precoders this lane's D rows cover
  float nv0[8], nv1[8];
  #pragma unroll
  for (int j = 0; j < 8; ++j) { nv0[j] = nvg[4*j + 2*h]; nv1[j] = nvg[4*j + 2*h + 1]; }

  float acc0[8], acc1[8];
  #pragma unroll
  for (int j = 0; j < 8; ++j) { acc0[j] = 0.f; acc1[j] = 0.f; }

  // 448 (batch,time) positions in 28 groups of 16 (N dimension)
  for (int g = 0; g < 28; ++g) {
    int bt = g*16 + r;
    const float* v = havg + ((size_t)bt * 256 + rb) * 16;
    if (g + 1 < 28)
      __builtin_prefetch(havg + ((size_t)(bt + 16) * 256 + rb) * 16, 0, 0);
    v2f bF[4];
    #pragma unroll
    for (int q = 0; q < 4; ++q) {       // B: lane n = column, K = 4q+2h+{0,1}
      int k = 4*q + 2*h;
      v2f bb; bb.x = v[k]; bb.y = v[k+1];
      bF[q] = bb;
    }
    #pragma unroll
    for (int j = 0; j < 8; ++j) {
      v8f d = {0.f,0.f,0.f,0.f,0.f,0.f,0.f,0.f};
      #pragma unroll
      for (int q = 0; q < 4; ++q)
        d = __builtin_amdgcn_wmma_f32_16x16x4_f32(false, aF[j][q],
                                                  false, bF[q],
                                                  (short)0, d, false, false);
      // lane holds rows 8h..8h+7 of column r: 2 precoders x (ReA,ImA,ReB,ImB)
      float p0 = d[0]*d[0] + d[1]*d[1] + d[2]*d[2] + d[3]*d[3];
      float x0 = 2.f*(d[0]*d[2] + d[1]*d[3]);
      acc0[j] += sinr_expterm(p0, x0, nv0[j]);
      float p1 = d[4]*d[4] + d[5]*d[5] + d[6]*d[6] + d[7]*d[7];
      float x1 = 2.f*(d[4]*d[6] + d[5]*d[7]);
      acc1[j] += sinr_expterm(p1, x1, nv1[j]);
    }
  }

  // gather per-precoder partial sums (16 position-columns) via LDS
  __shared__ float sm[NC][16];
  #pragma unroll
  for (int j = 0; j < 8; ++j) {
    sm[4*j + 2*h    ][r] = acc0[j];
    sm[4*j + 2*h + 1][r] = acc1[j];
  }
  __syncthreads();

  // lane l owns precoder c = l
  float S = 0.f;
  #pragma unroll
  for (int n = 0; n < 16; ++n) S += sm[l][n];
  float avg  = S * (1.f / (float)NBT);
  float asr  = -BETA_F * logf(avg);
  float rate = 0.83f * log2f(1.f + 0.73f * asr);

  // wave-wide argmax (ties -> lowest index, matching jnp.argmax)
  float best = rate; int bc = l;
  #pragma unroll
  for (int off = 16; off > 0; off >>= 1) {
    float orate = __shfl_down(best, off, 32);
    int   oc    = __shfl_down(bc,   off, 32);
    if (orate > best || (orate == best && oc < bc)) { best = orate; bc = oc; }
  }
  if (l == 0) {
    pmi[rb]  = bc;
    rsel[rb] = best;
    float wr[4], wi[4];
    codebook_w(bc, wr, wi);
    #pragma unroll
    for (int t = 0; t < 4; ++t) {       // complex64 layout: re,im interleaved
      pmat[rb*8 + 2*t + 0] = wr[t];
      pmat[rb*8 + 2*t + 1] = wi[t];
    }
  }
}

extern "C" void kernel_launch(void* const* d_in, const int* in_sizes, int n_in,
                              void* d_out, int out_size, void* d_ws, size_t ws_size,
                              hipStream_t stream) {
  const float* hr = (const float*)d_in[0];   // h_est_real (32,1,2,1,4,14,3072)
  const float* hi = (const float*)d_in[1];   // h_est_imag

  // workspace: havg (1,835,008 f32 = 7.0 MiB) | R (32 f32) | nv (32 f32)
  float* havg = (float*)d_ws;
  float* R    = havg + (size_t)NPOS * 16;
  float* nv   = R + 32;

  // outputs, concatenated in return order:
  //   PMI: 256 int32 | rate_for_selected: 256 f32 | precoding (1,256,4,1) c64
  char*  ob   = (char*)d_out;
  int*   pmi  = (int*)ob;
  float* rsel = (float*)(ob + 256 * sizeof(int));
  float* pmat = (float*)(ob + 256 * sizeof(int) + 256 * sizeof(float));

  k_zero  <<<1,    64, 0, stream>>>(R, 64);
  k_rb_avg<<<3584, 256, 0, stream>>>(hr, hi, havg);   // 32*2*4*14*256 threads
  k_rsum  <<<448,  256, 0, stream>>>(havg, R);        // 114,688 positions
  k_nv    <<<1,    32, 0, stream>>>(R, nv);
  k_rate  <<<NRB,  32, 0, stream>>>(havg, nv, pmi, rsel, pmat);
}